// Attention_87110526697915
// MI455X (gfx1250) — compile-verified
//
#include <hip/hip_runtime.h>
#include <math.h>

#define N_NODES 50000
#define N_EDGES 800000
#define D_FEAT 256
#define SMALL 25
#define QK_STRIDE 32
#define TOT_EDGES (N_EDGES + N_NODES)

typedef __attribute__((ext_vector_type(2))) float v2f;
typedef __attribute__((ext_vector_type(8))) float v8f;

// monotonic float <-> int map for exact integer atomicMax on floats
__device__ __forceinline__ int f32_to_ordered(float f) {
  int i = __float_as_int(f);
  return (i >= 0) ? i : (i ^ 0x7fffffff);
}
__device__ __forceinline__ float ordered_to_f32(int o) {
  int i = (o >= 0) ? o : (o ^ 0x7fffffff);
  return __int_as_float(i);
}

// ---------------------------------------------------------------------------
// 0) init: zero out[], denom[], set wmax sentinel
// ---------------------------------------------------------------------------
__global__ void __launch_bounds__(256) init_kernel(float* __restrict__ out,
                                                   int* __restrict__ wmax_ord,
                                                   float* __restrict__ denom) {
  long i = (long)blockIdx.x * blockDim.x + threadIdx.x;
  if (i < (long)N_NODES * D_FEAT) out[i] = 0.0f;
  if (i < N_NODES) {
    wmax_ord[i] = (int)0x80000000;  // below ordered(-inf)
    denom[i] = 0.0f;
  }
}

// ---------------------------------------------------------------------------
// 1) q = tanh(x@Wq + bq), k = x@Wk + bk via V_WMMA_F32_16X16X4_F32
//    Each wave: 16 node-rows x 32 padded cols (2 col tiles) for both q and k.
//    Block = 256 threads = 8 waves = 128 rows.
// ---------------------------------------------------------------------------
__global__ void __launch_bounds__(256) qk_gemm_kernel(
    const float* __restrict__ x, const float* __restrict__ Wq,
    const float* __restrict__ bq, const float* __restrict__ Wk,
    const float* __restrict__ bk, float* __restrict__ qpad,
    float* __restrict__ kpad) {
  // Transposed + zero-padded weights in LDS: sW[n][k], n in [0,32), k in [0,256)
  __shared__ float sW[2 * QK_STRIDE * D_FEAT];  // 64 KB
  float* sWq = sW;
  float* sWk = sW + QK_STRIDE * D_FEAT;

  const int tid = threadIdx.x;
  for (int idx = tid; idx < QK_STRIDE * D_FEAT; idx += 256) {
    int n = idx / D_FEAT;
    int k = idx - n * D_FEAT;
    float wq = 0.0f, wk = 0.0f;
    if (n < SMALL) {
      wq = Wq[k * SMALL + n];
      wk = Wk[k * SMALL + n];
    }
    sWq[idx] = wq;
    sWk[idx] = wk;
  }
  __syncthreads();

  const int lane = tid & 31;
  const int wave = tid >> 5;
  const int row0 = (blockIdx.x * 8 + wave) * 16;
  const int n = lane & 15;                 // column within 16-wide tile / A row
  const int kk0 = (lane < 16) ? 0 : 2;     // K sub-offset for both A and B frags

  int row = row0 + n;
  int rowc = (row < N_NODES) ? row : (N_NODES - 1);  // clamp loads, guard stores
  const float* xrow = x + (long)rowc * D_FEAT;
  const float* bq0p = sWq + (n + 0) * D_FEAT;
  const float* bq1p = sWq + (n + 16) * D_FEAT;
  const float* bk0p = sWk + (n + 0) * D_FEAT;
  const float* bk1p = sWk + (n + 16) * D_FEAT;

  v8f cq0 = {}, cq1 = {}, ck0 = {}, ck1 = {};
  for (int k0 = 0; k0 < D_FEAT; k0 += 4) {
    int ka = k0 + kk0;
    v2f a   = *(const v2f*)(xrow + ka);
    v2f bq0 = *(const v2f*)(bq0p + ka);
    v2f bq1 = *(const v2f*)(bq1p + ka);
    v2f bk0 = *(const v2f*)(bk0p + ka);
    v2f bk1 = *(const v2f*)(bk1p + ka);
    cq0 = __builtin_amdgcn_wmma_f32_16x16x4_f32(false, a, false, bq0, (short)0, cq0, false, false);
    cq1 = __builtin_amdgcn_wmma_f32_16x16x4_f32(false, a, false, bq1, (short)0, cq1, false, false);
    ck0 = __builtin_amdgcn_wmma_f32_16x16x4_f32(false, a, false, bk0, (short)0, ck0, false, false);
    ck1 = __builtin_amdgcn_wmma_f32_16x16x4_f32(false, a, false, bk1, (short)0, ck1, false, false);
  }

  // D layout: element i -> row = row0 + i + (lane<16 ? 0 : 8), col = tilebase + n
  const int rbase = row0 + ((lane < 16) ? 0 : 8);
  const float bq0s = (n < SMALL) ? bq[n] : 0.0f;
  const float bk0s = (n < SMALL) ? bk[n] : 0.0f;
  const bool hi_valid = (n + 16) < SMALL;
  const float bq1s = hi_valid ? bq[n + 16] : 0.0f;
  const float bk1s = hi_valid ? bk[n + 16] : 0.0f;

#pragma unroll
  for (int i = 0; i < 8; ++i) {
    int r = rbase + i;
    if (r < N_NODES) {
      long base = (long)r * QK_STRIDE;
      qpad[base + n]      = tanhf(cq0[i] + bq0s);
      qpad[base + n + 16] = hi_valid ? tanhf(cq1[i] + bq1s) : 0.0f;
      kpad[base + n]      = ck0[i] + bk0s;
      kpad[base + n + 16] = hi_valid ? (ck1[i] + bk1s) : 0.0f;
    }
  }
}

// ---------------------------------------------------------------------------
// 2) per-edge w = <q[s], k[d]> / 5 ; segment max via int atomicMax.
//    8 lanes per edge, float4 loads, shuffle reduction.
// ---------------------------------------------------------------------------
__global__ void __launch_bounds__(256) edge_w_kernel(
    const int* __restrict__ src, const int* __restrict__ dst,
    const float* __restrict__ qpad, const float* __restrict__ kpad,
    float* __restrict__ wbuf, int* __restrict__ wmax_ord) {
  const int tid = threadIdx.x;
  const int e = blockIdx.x * 32 + (tid >> 3);
  if (e >= TOT_EDGES) return;
  const int sub = tid & 7;

  int s, d;
  if (e < N_EDGES) {
    s = src[e];
    d = dst[e];
  } else {
    s = e - N_EDGES;
    d = s;
  }
  const float4 qv = *(const float4*)(qpad + (long)s * QK_STRIDE + sub * 4);
  const float4 kv = *(const float4*)(kpad + (long)d * QK_STRIDE + sub * 4);
  float acc = qv.x * kv.x + qv.y * kv.y + qv.z * kv.z + qv.w * kv.w;
  acc += __shfl_xor(acc, 1, 32);
  acc += __shfl_xor(acc, 2, 32);
  acc += __shfl_xor(acc, 4, 32);
  if (sub == 0) {
    float w = acc * 0.2f;  // 1/sqrt(25)
    wbuf[e] = w;
    atomicMax(wmax_ord + d, f32_to_ordered(w));
  }
}

// ---------------------------------------------------------------------------
// 3) ew = exp(w - wmax[d]); denom[d] += ew  (native f32 atomic add)
// ---------------------------------------------------------------------------
__global__ void __launch_bounds__(256) edge_exp_kernel(
    const int* __restrict__ dst, float* __restrict__ wbuf,
    const int* __restrict__ wmax_ord, float* __restrict__ denom) {
  int e = blockIdx.x * 256 + threadIdx.x;
  if (e >= TOT_EDGES) return;
  int d = (e < N_EDGES) ? dst[e] : (e - N_EDGES);
  float wmax = ordered_to_f32(wmax_ord[d]);
  float ew = expf(wbuf[e] - wmax);
  wbuf[e] = ew;  // recomputed every launch -> deterministic across replays
  __hip_atomic_fetch_add(denom + d, ew, __ATOMIC_RELAXED, __HIP_MEMORY_SCOPE_AGENT);
}

// ---------------------------------------------------------------------------
// 4) out[d,:] += (ew/denom[d]) * x[s,:]  — one wave per edge, 8 cols per lane
// ---------------------------------------------------------------------------
__global__ void __launch_bounds__(256) scatter_kernel(
    const float* __restrict__ x, const int* __restrict__ src,
    const int* __restrict__ dst, const float* __restrict__ ew,
    const float* __restrict__ denom, float* __restrict__ out) {
  const int lane = threadIdx.x & 31;
  const int e = blockIdx.x * 8 + (threadIdx.x >> 5);
  if (e >= TOT_EDGES) return;
  int s, d;
  if (e < N_EDGES) {
    s = src[e];
    d = dst[e];
  } else {
    s = e - N_EDGES;
    d = s;
  }
  const float alpha = ew[e] / denom[d];
  const float* xs = x + (long)s * D_FEAT;
  float* od = out + (long)d * D_FEAT;
#pragma unroll
  for (int i = 0; i < 8; ++i) {
    int c = lane + i * 32;
    __hip_atomic_fetch_add(od + c, alpha * xs[c], __ATOMIC_RELAXED,
                           __HIP_MEMORY_SCOPE_AGENT);
  }
}

// ---------------------------------------------------------------------------
extern "C" void kernel_launch(void* const* d_in, const int* in_sizes, int n_in,
                              void* d_out, int out_size, void* d_ws, size_t ws_size,
                              hipStream_t stream) {
  (void)in_sizes; (void)n_in; (void)out_size; (void)ws_size;
  const float* x  = (const float*)d_in[0];
  const int*   src = (const int*)d_in[1];
  const int*   dst = (const int*)d_in[2];
  const float* Wq = (const float*)d_in[3];
  const float* bq = (const float*)d_in[4];
  const float* Wk = (const float*)d_in[5];
  const float* bk = (const float*)d_in[6];
  float* out = (float*)d_out;

  // workspace layout (floats): qpad | kpad | wbuf | wmax_ord(int) | denom
  float* qpad  = (float*)d_ws;
  float* kpad  = qpad + (size_t)N_NODES * QK_STRIDE;
  float* wbuf  = kpad + (size_t)N_NODES * QK_STRIDE;
  int*   wmax  = (int*)(wbuf + TOT_EDGES);
  float* denom = (float*)(wmax + N_NODES);

  {
    long total = (long)N_NODES * D_FEAT;
    init_kernel<<<(int)((total + 255) / 256), 256, 0, stream>>>(out, wmax, denom);
  }
  qk_gemm_kernel<<<(N_NODES + 127) / 128, 256, 0, stream>>>(x, Wq, bq, Wk, bk,
                                                            qpad, kpad);
  edge_w_kernel<<<(TOT_EDGES + 31) / 32, 256, 0, stream>>>(src, dst, qpad, kpad,
                                                           wbuf, wmax);
  edge_exp_kernel<<<(TOT_EDGES + 255) / 256, 256, 0, stream>>>(dst, wbuf, wmax,
                                                               denom);
  scatter_kernel<<<(TOT_EDGES + 7) / 8, 256, 0, stream>>>(x, src, dst, wbuf,
                                                          denom, out);
}